// EulerianModelWrapper_68994354643550
// MI455X (gfx1250) — compile-verified
//
#include <hip/hip_runtime.h>

typedef __attribute__((ext_vector_type(2))) float v2f;
typedef __attribute__((ext_vector_type(8))) float v8f;

#define GRID_H   128
#define GRID_W   128
#define NVOX     (GRID_H * GRID_W)
#define NTHREADS 256
#define NWAVES   (NTHREADS / 32)
#define XMIN     (-0.3f)
#define XSTEP    (0.6f / 127.0f)
#define ZMID     0.745f

__device__ __forceinline__ float vcoord(int i) { return XMIN + (float)i * XSTEP; }

// Raw v_min_num_f32: skips the v_max_num canonicalize pair fminf() emits.
// Inputs are squared distances or -inf -- never sNaN -- so this is safe.
__device__ __forceinline__ float min_raw(float a, float b) {
    float r;
    asm("v_min_num_f32 %0, %1, %2" : "=v"(r) : "v"(a), "v"(b));
    return r;
}

// One workgroup per batch. 8 waves; wave w owns the 16-row tile strip tr=w.
// Distance field lives in VGPRs in WMMA accumulator layout:
//   dist[tc][r] holds voxel (i = tr*16 + r + 8*(lane>>4), j = tc*16 + (lane&15)).
__global__ __launch_bounds__(NTHREADS, 2)
void fps_pipeline_kernel(const float* __restrict__ s_cur,
                         const float* __restrict__ action,
                         const float* __restrict__ camE,
                         const float* __restrict__ wact,
                         int n_pts, int n_particles,
                         float* __restrict__ out)
{
    __shared__ unsigned occBits[NVOX / 32];   // 2 KB occupancy bitmask
    __shared__ float    sBias;
    __shared__ int      sAny;
    __shared__ int      sStart;
    __shared__ float    redVal[2][NWAVES];    // double-buffered: 1 barrier/round
    __shared__ int      redIdx[2][NWAVES];

    const int b    = blockIdx.x;
    const int tid  = threadIdx.x;
    const int lane = tid & 31;
    const int wid  = tid >> 5;
    const float NEG_INF = __int_as_float(0xff800000);

    // ---- phase 0: zero bitmask; thread 0 computes action-conditioned bias ----
    for (int w = tid; w < NVOX / 32; w += NTHREADS) occBits[w] = 0u;
    if (tid == 0) {
        // M = inv(inv(E) @ T) = T @ E  (T = diag(1,-1,-1,1) is self-inverse)
        const float a0 = action[b * 4 + 0], a1 = action[b * 4 + 1];
        const float a2 = action[b * 4 + 2], a3 = action[b * 4 + 3];
        float bias = 0.f;
        #pragma unroll
        for (int c = 0; c < 3; ++c) {
            const float sgn = (c == 1 || c == 2) ? -1.f : 1.f;
            const float e0 = camE[c * 4 + 0], e2 = camE[c * 4 + 2], e3 = camE[c * 4 + 3];
            const float s3  = sgn * (e0 * a0 - e2 * a1 + e3) * (1.f / 24.f);
            const float e3d = sgn * (e0 * a2 - e2 * a3 + e3) * (1.f / 24.f);
            bias += s3 * wact[c] + e3d * wact[c + 3];
        }
        sBias  = bias;
        sAny   = 0;
        sStart = 0x7fffffff;
    }
    __syncthreads();

    // ---- phase 1: particle -> occupancy-bit scatter (LDS atomics) ----
    const float* __restrict__ sp = s_cur + (size_t)b * n_pts * 3;
    for (int n = tid; n < n_pts; n += NTHREADS) {
        const float px = sp[n * 3 + 0];
        const float py = sp[n * 3 + 1];
        int ix = (int)rintf((px - XMIN) * (127.f / 0.6f));
        int iy = (int)rintf((py - XMIN) * (127.f / 0.6f));
        ix = ix < 0 ? 0 : (ix > GRID_H - 1 ? GRID_H - 1 : ix);
        iy = iy < 0 ? 0 : (iy > GRID_W - 1 ? GRID_W - 1 : iy);
        const int flat = ix * GRID_W + iy;
        atomicOr(&occBits[flat >> 5], 1u << (flat & 31));
    }
    __syncthreads();

    // ---- phase 2: mask predicate (occ is 0/1 after clip -> pure bit test) ----
    unsigned myor = 0u;
    for (int w = tid; w < NVOX / 32; w += NTHREADS) myor |= occBits[w];
    if (myor) atomicOr(&sAny, 1);
    __syncthreads();

    const float bias    = sBias;
    const bool  pred1   = (bias >= -2.f);  // sigmoid(4*1-2+bias) >= 0.5
    const bool  pred0   = (bias >=  2.f);  // sigmoid(4*0-2+bias) >= 0.5
    const bool  anyPred = (sAny && pred1) || pred0;
    const bool  fallAll = !anyPred;        // empty mask -> all voxels

    // ---- phase 3: deterministic seed = first masked voxel in flat order ----
    {
        int firstLocal = 0x7fffffff;
        const int base = tid * (NVOX / NTHREADS);
        for (int k = 0; k < NVOX / NTHREADS; ++k) {
            const int v = base + k;
            const bool occ = (occBits[v >> 5] >> (v & 31)) & 1u;
            if (fallAll || (occ ? pred1 : pred0)) { firstLocal = v; break; }
        }
        atomicMin(&sStart, firstLocal);
    }
    __syncthreads();

    const int   startV = sStart;
    const float spx = vcoord(startV >> 7);
    const float spy = vcoord(startV & (GRID_W - 1));
    if (tid == 0) {
        float* o = out + (size_t)b * n_particles * 3;
        o[0] = spx; o[1] = spy; o[2] = ZMID;
    }

    // ---- phase 4: init VGPR-resident distance field (squared dist; -inf = unmasked) ----
    const int tr    = wid;
    const int Mbase = 8 * (lane >> 4);
    const int Ncol  = lane & 15;
    const int vBase = (tr * 16 + Mbase) * GRID_W + Ncol;

    v8f dist[8];
    #pragma unroll
    for (int tc = 0; tc < 8; ++tc) {
        #pragma unroll
        for (int r = 0; r < 8; ++r) {
            const int i = tr * 16 + Mbase + r;
            const int j = tc * 16 + Ncol;
            const int v = i * GRID_W + j;
            const bool occ = (occBits[v >> 5] >> (v & 31)) & 1u;
            const bool m = fallAll || (occ ? pred1 : pred0);
            const float dx = vcoord(i) - spx;
            const float dy = vcoord(j) - spy;
            dist[tc][r] = m ? (dx * dx + dy * dy) : NEG_INF;
        }
    }

    // ---- phase 5: farthest-point sampling loop (one barrier per round) ----
    for (int t = 1; t < n_particles; ++t) {
        const int pb = t & 1;

        // per-lane argmax (first-index tie break, matching jnp.argmax)
        float bv = NEG_INF;
        int   bi = 0x7fffffff;
        #pragma unroll
        for (int tc = 0; tc < 8; ++tc) {
            #pragma unroll
            for (int r = 0; r < 8; ++r) {
                const float d = dist[tc][r];
                const int v = vBase + r * GRID_W + tc * 16;
                if (d > bv || (d == bv && v < bi)) { bv = d; bi = v; }
            }
        }
        // wave32 butterfly reduction
        #pragma unroll
        for (int off = 16; off > 0; off >>= 1) {
            const float ov = __shfl_xor(bv, off, 32);
            const int   oi = __shfl_xor(bi, off, 32);
            if (ov > bv || (ov == bv && oi < bi)) { bv = ov; bi = oi; }
        }
        if (lane == 0) { redVal[pb][wid] = bv; redIdx[pb][wid] = bi; }
        __syncthreads();

        // all threads redundantly reduce the 8 wave candidates (LDS broadcast
        // reads) -- removes the serialized thread-0 reduce + second barrier.
        float gv = redVal[pb][0];
        int   gi = redIdx[pb][0];
        #pragma unroll
        for (int w = 1; w < NWAVES; ++w) {
            const float wv = redVal[pb][w];
            const int   wi = redIdx[pb][w];
            if (wv > gv || (wv == gv && wi < gi)) { gv = wv; gi = wi; }
        }
        const float fx = vcoord(gi >> 7);
        const float fy = vcoord(gi & (GRID_W - 1));
        if (tid == 0) {
            float* o = out + ((size_t)b * n_particles + t) * 3;
            o[0] = fx; o[1] = fy; o[2] = ZMID;
        }

        // rank-2 separable distance tile via WMMA:
        //   D[m][n] = a_m*1 + 1*b_n  with a_m=(x_i-fx)^2, b_n=(y_j-fy)^2
        // A (16x4): lanes 0-15 hold K=0,1 for row M=lane; lanes 16-31 (K=2,3) zero.
        const float axv = (lane < 16) ? (vcoord(tr * 16 + lane) - fx) : 0.f;
        v2f A;
        A.x = axv * axv;                    // K=0 column: a_m
        A.y = (lane < 16) ? 1.f : 0.f;      // K=1 column: ones
        const v8f zero = {};
        #pragma unroll
        for (int tc = 0; tc < 8; ++tc) {
            const float byv = (lane < 16) ? (vcoord(tc * 16 + lane) - fy) : 0.f;
            v2f Bm;
            Bm.x = (lane < 16) ? 1.f : 0.f; // K=0 row: ones
            Bm.y = byv * byv;               // K=1 row: b_n
            const v8f nd = __builtin_amdgcn_wmma_f32_16x16x4_f32(
                /*neg_a=*/false, A, /*neg_b=*/false, Bm,
                /*c_mod=*/(short)0, zero, /*reuse_a=*/false, /*reuse_b=*/false);
            #pragma unroll
            for (int r = 0; r < 8; ++r)
                dist[tc][r] = min_raw(dist[tc][r], nd[r]);  // -inf absorbs (mask)
        }
    }
}

extern "C" void kernel_launch(void* const* d_in, const int* in_sizes, int n_in,
                              void* d_out, int out_size, void* d_ws, size_t ws_size,
                              hipStream_t stream) {
    const float* s_cur  = (const float*)d_in[0];
    const float* action = (const float*)d_in[1];
    const float* camE   = (const float*)d_in[2];
    const float* wact   = (const float*)d_in[3];

    const int B      = in_sizes[1] / 4;            // action is (B,4)
    const int n_pts  = in_sizes[0] / (B * 3);      // s_cur is (B,N,3)
    const int n_part = out_size / (B * 3);         // out is (B,n_particles,3)

    fps_pipeline_kernel<<<dim3(B), dim3(NTHREADS), 0, stream>>>(
        s_cur, action, camE, wact, n_pts, n_part, (float*)d_out);
}